// MyParticleNetwork_61323543052534
// MI455X (gfx1250) — compile-verified
//
#include <hip/hip_runtime.h>
#include <math.h>

// ---------------------------------------------------------------------------
// MI455X (gfx1250) particle-network forward pass.
// wave32; V_WMMA_F32_16X16X4_F32 for all GEMMs + flash attention;
// 262KB-LDS fused ContinuousConv tiles (only possible with CDNA5's 320KB WGP LDS).
// ---------------------------------------------------------------------------

typedef __attribute__((ext_vector_type(2))) float v2f;
typedef __attribute__((ext_vector_type(8))) float v8f;

#define DT        0.02f
#define NQ        16      // queries per cconv workgroup (one WMMA M-tile)
#define NBRS      32      // neighbors per query

__device__ __forceinline__ float signf(float x) {
  return (x > 0.f) ? 1.f : ((x < 0.f) ? -1.f : 0.f);
}

// reduction across the 16 lanes of a half-wave (matches WMMA C-fragment rows)
__device__ __forceinline__ float half_max(float v) {
  v = fmaxf(v, __shfl_xor(v, 1, 32));
  v = fmaxf(v, __shfl_xor(v, 2, 32));
  v = fmaxf(v, __shfl_xor(v, 4, 32));
  v = fmaxf(v, __shfl_xor(v, 8, 32));
  return v;
}
__device__ __forceinline__ float half_sum(float v) {
  v += __shfl_xor(v, 1, 32);
  v += __shfl_xor(v, 2, 32);
  v += __shfl_xor(v, 4, 32);
  v += __shfl_xor(v, 8, 32);
  return v;
}
__device__ __forceinline__ float wave_sum(float v) {
#pragma unroll
  for (int off = 16; off >= 1; off >>= 1) v += __shfl_xor(v, off, 32);
  return v;
}

// Open3D ball_to_cube_volume_preserving (sphere->cylinder->cube), fp32.
__device__ __forceinline__ void ball_to_cube(float x, float y, float z,
                                             float& U, float& V, float& W) {
  const float eps = 1e-8f;
  float sq    = x * x + y * y + z * z;
  float nrm   = sqrtf(sq);
  float sq_xy = x * x + y * y;
  bool  polar = (1.25f * z * z) > sq_xy;
  float sa = sqrtf(3.f * nrm / (nrm + fabsf(z) + eps));
  float sb = nrm / (sqrtf(sq_xy) + eps);
  float xc = polar ? x * sa : x * sb;
  float yc = polar ? y * sa : y * sb;
  float zc = polar ? signf(z) * nrm : 1.5f * z;
  if (sq < eps) { xc = x; yc = y; zc = z; }
  float nxy = sqrtf(xc * xc + yc * yc);
  const float fo_pi = 1.27323954473516277f;  // 4/pi
  float sx = (fabsf(xc) > eps) ? xc : eps;
  float sy = (fabsf(yc) > eps) ? yc : eps;
  float ux = signf(xc) * nxy;
  float vy = signf(yc) * nxy;
  bool xbig = fabsf(yc) <= fabsf(xc);
  float u = xbig ? ux : vy * fo_pi * atanf(xc / sy);
  float v = xbig ? ux * fo_pi * atanf(yc / sx) : vy;
  if (nxy < eps) { u = xc; v = yc; }
  U = u; V = v; W = zc;
}

// ---------------------------------------------------------------------------
// K0: integrate + fluid feature assembly [1, vel2]
// ---------------------------------------------------------------------------
__global__ void integrate_kernel(const float* __restrict__ pos,
                                 const float* __restrict__ vel,
                                 float* __restrict__ pos2,
                                 float* __restrict__ feats, int N) {
  int i = blockIdx.x * blockDim.x + threadIdx.x;
  if (i >= N) return;
  float vx = vel[i * 3 + 0], vy = vel[i * 3 + 1], vz = vel[i * 3 + 2];
  float v2x = vx, v2y = vy - 9.81f * DT, v2z = vz;
  pos2[i * 3 + 0] = pos[i * 3 + 0] + DT * 0.5f * (v2x + vx);
  pos2[i * 3 + 1] = pos[i * 3 + 1] + DT * 0.5f * (v2y + vy);
  pos2[i * 3 + 2] = pos[i * 3 + 2] + DT * 0.5f * (v2z + vz);
  feats[i * 4 + 0] = 1.f;
  feats[i * 4 + 1] = v2x;
  feats[i * 4 + 2] = v2y;
  feats[i * 4 + 3] = v2z;
}

// ---------------------------------------------------------------------------
// Fused ContinuousConv: per-16-query tile, scatter trilinear bins into LDS
// (stride padded for bank-conflict-free access), then 16xCout = B(16xK) @ W
// with V_WMMA_F32_16X16X4_F32.  K = 64*CIN (bins x in-channels).
// ---------------------------------------------------------------------------
template <int CIN, int NT>
__global__ void __launch_bounds__(32 * NT)
cconv_fused(const float* __restrict__ fsrc, const float* __restrict__ pos2,
            const int* __restrict__ nbrs, const float* __restrict__ Wc,
            const float* __restrict__ bias, float* __restrict__ out, int Cout) {
  constexpr int K      = 64 * CIN;
  constexpr int STRIDE = K + 4;   // +4 dwords: lane m spread over banks 4m%64
  extern __shared__ float smem[]; // NQ * STRIDE floats
  const int tid   = threadIdx.x;
  const int nth   = 32 * NT;
  const int qbase = blockIdx.x * NQ;
  const float INV_R = 1.0f / (0.5f * (1.5f * 6.0f * 0.025f)); // 1/0.1125

  for (int i = tid; i < NQ * STRIDE / 4; i += nth)
    ((float4*)smem)[i] = float4{0.f, 0.f, 0.f, 0.f};
  __syncthreads();

  // ---- bin build: lane->query mapping keeps lanes on distinct LDS banks ----
  {
    const int q = tid & 15;
    const int n = qbase + q;
    const float px = pos2[n * 3 + 0];
    const float py = pos2[n * 3 + 1];
    const float pz = pos2[n * 3 + 2];
    for (int j = tid >> 4; j < NBRS; j += (nth >> 4)) {
      const int nb = nbrs[n * NBRS + j];
      float rx = (pos2[nb * 3 + 0] - px) * INV_R;
      float ry = (pos2[nb * 3 + 1] - py) * INV_R;
      float rz = (pos2[nb * 3 + 2] - pz) * INV_R;
      float r2 = rx * rx + ry * ry + rz * rz;
      float w = 0.f;
      if (r2 < 1.f) { float t = 1.f - r2; w = t * t * t; }
      if (nb == n) w = 0.f;          // radius_search_ignore_query_points
      if (w == 0.f) continue;        // zero contribution
      float bu, bv, bw;
      ball_to_cube(rx, ry, rz, bu, bv, bw);
      float tx = fminf(fmaxf((bu + 1.f) * 1.5f, 0.f), 3.f);
      float ty = fminf(fmaxf((bv + 1.f) * 1.5f, 0.f), 3.f);
      float tz = fminf(fmaxf((bw + 1.f) * 1.5f, 0.f), 3.f);
      int ix = (int)floorf(tx); ix = ix < 0 ? 0 : (ix > 2 ? 2 : ix);
      int iy = (int)floorf(ty); iy = iy < 0 ? 0 : (iy > 2 ? 2 : iy);
      int iz = (int)floorf(tz); iz = iz < 0 ? 0 : (iz > 2 ? 2 : iz);
      float fx = tx - (float)ix, fy = ty - (float)iy, fz = tz - (float)iz;

      float fv[CIN];
#pragma unroll
      for (int c = 0; c < CIN; ++c) fv[c] = w * fsrc[nb * CIN + c];

      float* qrow = smem + q * STRIDE;
#pragma unroll
      for (int cx = 0; cx < 2; ++cx) {
        float wx = cx ? fx : (1.f - fx);
#pragma unroll
        for (int cy = 0; cy < 2; ++cy) {
          float wxy = wx * (cy ? fy : (1.f - fy));
#pragma unroll
          for (int cz = 0; cz < 2; ++cz) {
            float cw = wxy * (cz ? fz : (1.f - fz));
            int bin = (ix + cx) * 16 + (iy + cy) * 4 + (iz + cz);
            float* dst = qrow + bin * CIN;
#pragma unroll
            for (int c = 0; c < CIN; ++c) atomicAdd(dst + c, cw * fv[c]);
          }
        }
      }
    }
  }
  __syncthreads();

  // ---- WMMA: one 16x16 output tile per wave, K-loop over LDS bins ----------
  const int lane  = tid & 31;
  const int wv    = tid >> 5;       // output column tile
  const int half  = lane >> 4;      // K-half select (ISA f32 A/B layout)
  const int mrow  = lane & 15;      // A row
  const int col   = wv * 16 + (lane & 15);
  const bool colok = col < Cout;
  const int  colc  = colok ? col : (Cout - 1);   // clamped: loads stay uniform
  const float cmask = colok ? 1.f : 0.f;
  v8f acc = {0.f, 0.f, 0.f, 0.f, 0.f, 0.f, 0.f, 0.f};
#pragma unroll 4
  for (int k = 0; k < K; k += 4) {
    const int ka = k + half * 2;
    v2f a, b;
    a[0] = smem[mrow * STRIDE + ka + 0];
    a[1] = smem[mrow * STRIDE + ka + 1];
    b[0] = Wc[(ka + 0) * Cout + colc] * cmask;   // v_cndmask/mul, no exec branch
    b[1] = Wc[(ka + 1) * Cout + colc] * cmask;
    acc = __builtin_amdgcn_wmma_f32_16x16x4_f32(false, a, false, b,
                                                (short)0, acc, false, false);
  }
  if (colok) {
    float bb = bias[col];
#pragma unroll
    for (int r = 0; r < 8; ++r) {
      int m = r + half * 8;
      out[(qbase + m) * Cout + col] = acc[r] + bb;
    }
  }
}

// ---------------------------------------------------------------------------
// Generic fp32 WMMA GEMM: C = [C +] A(MxK) @ B(KxN) + bias [+ addend] [relu]
// one 16x16 tile per wave; M multiple of 16, K multiple of 4.
// flags: 1 = accumulate into existing C, 2 = relu output.
// ---------------------------------------------------------------------------
__global__ void __launch_bounds__(32)
gemm_wmma(const float* __restrict__ A, int lda,
          const float* __restrict__ B, int ldb,
          const float* __restrict__ bias,
          const float* __restrict__ addend,
          float* __restrict__ C, int ldc,
          int Ncols, int K, int flags) {
  const int lane = threadIdx.x & 31;
  const int half = lane >> 4;
  const int mrow = (blockIdx.y << 4) + (lane & 15);
  const int col  = (blockIdx.x << 4) + (lane & 15);
  const bool colok = col < Ncols;
  const int  colc  = colok ? col : (Ncols - 1);
  const float cmask = colok ? 1.f : 0.f;
  v8f acc = {0.f, 0.f, 0.f, 0.f, 0.f, 0.f, 0.f, 0.f};
#pragma unroll 8
  for (int k = 0; k < K; k += 4) {
    const int ka = k + half * 2;
    v2f a, b;
    a[0] = A[mrow * lda + ka + 0];
    a[1] = A[mrow * lda + ka + 1];
    b[0] = B[(ka + 0) * ldb + colc] * cmask;
    b[1] = B[(ka + 1) * ldb + colc] * cmask;
    acc = __builtin_amdgcn_wmma_f32_16x16x4_f32(false, a, false, b,
                                                (short)0, acc, false, false);
  }
  if (!colok) return;
  const float bb = bias ? bias[col] : 0.f;
#pragma unroll
  for (int r = 0; r < 8; ++r) {
    int m = (blockIdx.y << 4) + r + half * 8;
    float v = acc[r] + bb;
    if (addend) v += addend[m * ldc + col];
    if (flags & 1) v += C[m * ldc + col];
    if (flags & 2) v = fmaxf(v, 0.f);
    C[m * ldc + col] = v;
  }
}

// ---------------------------------------------------------------------------
// LayerNorm over 32 channels (torch-style: unbiased std, eps outside sqrt)
// ---------------------------------------------------------------------------
__global__ void __launch_bounds__(32)
layernorm32(const float* __restrict__ x, const float* __restrict__ a,
            const float* __restrict__ b, float* __restrict__ y) {
  const int n = blockIdx.x, lane = threadIdx.x;
  float v  = x[n * 32 + lane];
  float mu = wave_sum(v) * (1.f / 32.f);
  float d  = v - mu;
  float sd = sqrtf(wave_sum(d * d) * (1.f / 31.f));
  y[n * 32 + lane] = a[lane] * d / (sd + 1e-6f) + b[lane];
}

// ---------------------------------------------------------------------------
// WMMA flash attention, h=1, d=32.  One wave per 16-query tile; 16-key tiles.
// S = Q@K^T via 8 f32-WMMA steps, online softmax on the C-layout fragments
// (row stats = 16-lane half-wave reductions), P re-staged C->A layout through
// a bank-conflict-free LDS tile (stride 20), O += P@V via 8 WMMA steps.
// ---------------------------------------------------------------------------
__global__ void __launch_bounds__(32)
attn_wmma(const float* __restrict__ Q, const float* __restrict__ Kmat,
          const float* __restrict__ V, float* __restrict__ O, int N) {
  __shared__ float plds[16 * 20];
  const int lane  = threadIdx.x;
  const int half  = lane >> 4;
  const int idx   = lane & 15;
  const int qbase = blockIdx.x * 16;
  const float scale = 0.17677669529663689f;  // 1/sqrt(32)

  // preload Q A-fragments (16 rows x 32 dims, 8 K-steps)
  v2f qa[8];
#pragma unroll
  for (int kk = 0; kk < 8; ++kk) {
    const float* qr = Q + (qbase + idx) * 32 + kk * 4 + half * 2;
    qa[kk][0] = qr[0];
    qa[kk][1] = qr[1];
  }
  float mrun[8], lrun[8];
#pragma unroll
  for (int r = 0; r < 8; ++r) { mrun[r] = -3.0e38f; lrun[r] = 0.f; }
  v8f o0 = {0.f, 0.f, 0.f, 0.f, 0.f, 0.f, 0.f, 0.f};
  v8f o1 = {0.f, 0.f, 0.f, 0.f, 0.f, 0.f, 0.f, 0.f};

  for (int kb = 0; kb < N; kb += 16) {
    // ---- S-tile = Q @ K^T  (B[k][n] = K[kb+n][k]) ----
    v8f s = {0.f, 0.f, 0.f, 0.f, 0.f, 0.f, 0.f, 0.f};
#pragma unroll
    for (int kk = 0; kk < 8; ++kk) {
      const float* kr = Kmat + (kb + idx) * 32 + kk * 4 + half * 2;
      v2f b; b[0] = kr[0]; b[1] = kr[1];
      s = __builtin_amdgcn_wmma_f32_16x16x4_f32(false, qa[kk], false, b,
                                                (short)0, s, false, false);
    }
    // ---- online softmax update (rows live in 16-lane halves) ----
    float p[8];
#pragma unroll
    for (int r = 0; r < 8; ++r) {
      float sv = s[r] * scale;
      float tm = half_max(sv);
      float mn = fmaxf(mrun[r], tm);
      float al = expf(mrun[r] - mn);
      float pv = expf(sv - mn);
      lrun[r] = lrun[r] * al + half_sum(pv);
      mrun[r] = mn;
      p[r]    = pv;
      o0[r] *= al;
      o1[r] *= al;
    }
    // ---- stage P: C-layout -> A-layout via LDS (stride 20, conflict-free) --
    __syncthreads();   // protect against prior-iteration readers (1 wave: cheap)
#pragma unroll
    for (int r = 0; r < 8; ++r) plds[(r + 8 * half) * 20 + idx] = p[r];
    __syncthreads();
    v2f pa[4];
#pragma unroll
    for (int kk = 0; kk < 4; ++kk) {
      pa[kk][0] = plds[idx * 20 + kk * 4 + half * 2 + 0];
      pa[kk][1] = plds[idx * 20 + kk * 4 + half * 2 + 1];
    }
    // ---- O += P @ V ----
#pragma unroll
    for (int kk = 0; kk < 4; ++kk) {
      const int krow = kb + kk * 4 + half * 2;
      v2f b0, b1;
      b0[0] = V[(krow + 0) * 32 + idx];
      b0[1] = V[(krow + 1) * 32 + idx];
      b1[0] = V[(krow + 0) * 32 + 16 + idx];
      b1[1] = V[(krow + 1) * 32 + 16 + idx];
      o0 = __builtin_amdgcn_wmma_f32_16x16x4_f32(false, pa[kk], false, b0,
                                                 (short)0, o0, false, false);
      o1 = __builtin_amdgcn_wmma_f32_16x16x4_f32(false, pa[kk], false, b1,
                                                 (short)0, o1, false, false);
    }
  }
  // ---- normalize + write ----
#pragma unroll
  for (int r = 0; r < 8; ++r) {
    const int m = qbase + r + 8 * half;
    const float inv = 1.f / lrun[r];
    O[m * 32 + idx]      = o0[r] * inv;
    O[m * 32 + 16 + idx] = o1[r] * inv;
  }
}

// ---------------------------------------------------------------------------
// Elementwise helpers
// ---------------------------------------------------------------------------
__global__ void relu_kernel(const float* __restrict__ x, float* __restrict__ y,
                            int n) {
  int i = blockIdx.x * blockDim.x + threadIdx.x;
  if (i < n) y[i] = fmaxf(x[i], 0.f);
}

__global__ void finalize_kernel(const float* __restrict__ pos,
                                const float* __restrict__ pos2,
                                const float* __restrict__ corr3,
                                float* __restrict__ out, int N) {
  int i = blockIdx.x * blockDim.x + threadIdx.x;
  if (i >= N * 3) return;
  float p = pos2[i] + corr3[i] * (1.f / 128.f);
  out[i] = p;                                   // pos_out
  out[N * 3 + i] = (p - pos[i]) * (1.f / DT);   // vel_out
}

// ---------------------------------------------------------------------------
extern "C" void kernel_launch(void* const* d_in, const int* in_sizes, int n_in,
                              void* d_out, int out_size, void* d_ws,
                              size_t ws_size, hipStream_t stream) {
  const float* pos   = (const float*)d_in[0];
  const float* vel   = (const float*)d_in[1];
  const int*   fnbrs = (const int*)d_in[4];
  const float* cw0f  = (const float*)d_in[6];
  const float* cb0f  = (const float*)d_in[7];
  // d_in[2,3,5,8,9]: box branch — result provably unused downstream; skipped.
  const float* cw1 = (const float*)d_in[10]; const float* cb1 = (const float*)d_in[11];
  const float* cw2 = (const float*)d_in[12]; const float* cb2 = (const float*)d_in[13];
  const float* cw3 = (const float*)d_in[14]; const float* cb3 = (const float*)d_in[15];
  const float* dw1 = (const float*)d_in[16]; const float* db1 = (const float*)d_in[17];
  const float* dw2 = (const float*)d_in[18]; const float* db2 = (const float*)d_in[19];
  const float* dw3 = (const float*)d_in[20]; const float* db3 = (const float*)d_in[21];
  const float* selfW = (const float*)d_in[22]; const float* selfb = (const float*)d_in[23];
  const float* srcW  = (const float*)d_in[24]; const float* srcb  = (const float*)d_in[25];
  const float* lna = (const float*)d_in[26]; const float* lnb = (const float*)d_in[27];
  const float* ffw1 = (const float*)d_in[28]; const float* ffb1 = (const float*)d_in[29];
  const float* ffw2 = (const float*)d_in[30]; const float* ffb2 = (const float*)d_in[31];

  const int N = in_sizes[0] / 3;  // 8192

  // workspace partition (floats)
  float* w = (float*)d_ws;
  float* pos2   = w; w += N * 3;
  float* feats4 = w; w += N * 4;
  float* a0f    = w; w += N * 32;
  float* nbuf   = w; w += N * 32;
  float* Qb     = w; w += N * 32;
  float* Kb     = w; w += N * 32;
  float* Vb     = w; w += N * 32;
  float* aO     = w; w += N * 32;
  float* h1     = w; w += N * 32;
  float* h2     = w; w += N * 32;
  float* t1     = w; w += N * 64;
  float* xa     = w; w += N * 64;
  float* xb     = w; w += N * 64;
  float* rx     = w; w += N * 64;
  float* x3     = w; w += N * 3;

  dim3 b256(256);
  dim3 gN((N + 255) / 256);

  // 1) integrate
  integrate_kernel<<<gN, b256, 0, stream>>>(pos, vel, pos2, feats4, N);

  // 2) a0f = cconv0(feats4), Cin=4 Cout=32, exclude_self
  cconv_fused<4, 2><<<N / NQ, 64, NQ * (64 * 4 + 4) * sizeof(float), stream>>>(
      feats4, pos2, fnbrs, cw0f, cb0f, a0f, 32);

  // 3) self-attention block: h1 = a0f + MHA(LN0(a0f))
  layernorm32<<<N, 32, 0, stream>>>(a0f, lna + 0, lnb + 0, nbuf);
  dim3 g32(2, N / 16);
  gemm_wmma<<<g32, 32, 0, stream>>>(nbuf, 32, selfW + 0 * 1024, 32, selfb + 0, nullptr, Qb, 32, 32, 32, 0);
  gemm_wmma<<<g32, 32, 0, stream>>>(nbuf, 32, selfW + 1 * 1024, 32, selfb + 32, nullptr, Kb, 32, 32, 32, 0);
  gemm_wmma<<<g32, 32, 0, stream>>>(nbuf, 32, selfW + 2 * 1024, 32, selfb + 64, nullptr, Vb, 32, 32, 32, 0);
  attn_wmma<<<N / 16, 32, 0, stream>>>(Qb, Kb, Vb, aO, N);
  gemm_wmma<<<g32, 32, 0, stream>>>(aO, 32, selfW + 3 * 1024, 32, selfb + 96, a0f, h1, 32, 32, 32, 0);

  // 4) source-attention block: h2 = a0f + MHA(LN1(a0f) -> q, h1 -> kv)
  layernorm32<<<N, 32, 0, stream>>>(a0f, lna + 32, lnb + 32, nbuf);
  gemm_wmma<<<g32, 32, 0, stream>>>(nbuf, 32, srcW + 0 * 1024, 32, srcb + 0, nullptr, Qb, 32, 32, 32, 0);
  gemm_wmma<<<g32, 32, 0, stream>>>(h1, 32, srcW + 1 * 1024, 32, srcb + 32, nullptr, Kb, 32, 32, 32, 0);
  gemm_wmma<<<g32, 32, 0, stream>>>(h1, 32, srcW + 2 * 1024, 32, srcb + 64, nullptr, Vb, 32, 32, 32, 0);
  attn_wmma<<<N / 16, 32, 0, stream>>>(Qb, Kb, Vb, aO, N);
  gemm_wmma<<<g32, 32, 0, stream>>>(aO, 32, srcW + 3 * 1024, 32, srcb + 96, a0f, h2, 32, 32, 32, 0);

  // 5) FFN: x = relu(LN2(h2) @ ffw1 + ffb1) @ ffw2 + ffb2   -> xa [N,64]
  layernorm32<<<N, 32, 0, stream>>>(h2, lna + 64, lnb + 64, nbuf);
  dim3 g64(4, N / 16);
  gemm_wmma<<<g64, 32, 0, stream>>>(nbuf, 32, ffw1, 64, ffb1, nullptr, t1, 64, 64, 32, 2);
  gemm_wmma<<<g64, 32, 0, stream>>>(t1, 64, ffw2, 64, ffb2, nullptr, xa, 64, 64, 64, 0);

  const size_t lds64 = NQ * (64 * 64 + 4) * sizeof(float);  // 262,400 B
  dim3 gNe((N * 64 + 255) / 256);

  // 6) conv layer 1 (residual): xb = cconv(relu(xa)) + relu(xa)@dw1+db1 + xa
  relu_kernel<<<gNe, b256, 0, stream>>>(xa, rx, N * 64);
  cconv_fused<64, 4><<<N / NQ, 128, lds64, stream>>>(rx, pos2, fnbrs, cw1, cb1, xb, 64);
  gemm_wmma<<<g64, 32, 0, stream>>>(rx, 64, dw1, 64, db1, xa, xb, 64, 64, 64, 1);

  // 7) conv layer 2 (residual): xa = cconv(relu(xb)) + relu(xb)@dw2+db2 + xb
  relu_kernel<<<gNe, b256, 0, stream>>>(xb, rx, N * 64);
  cconv_fused<64, 4><<<N / NQ, 128, lds64, stream>>>(rx, pos2, fnbrs, cw2, cb2, xa, 64);
  gemm_wmma<<<g64, 32, 0, stream>>>(rx, 64, dw2, 64, db2, xb, xa, 64, 64, 64, 1);

  // 8) conv layer 3 (no residual, Cout=3): x3 = cconv(relu(xa)) + relu(xa)@dw3+db3
  relu_kernel<<<gNe, b256, 0, stream>>>(xa, rx, N * 64);
  cconv_fused<64, 1><<<N / NQ, 32, lds64, stream>>>(rx, pos2, fnbrs, cw3, cb3, x3, 3);
  dim3 g3(1, N / 16);
  gemm_wmma<<<g3, 32, 0, stream>>>(rx, 64, dw3, 3, db3, nullptr, x3, 3, 3, 64, 1);

  // 9) pos_out / vel_out -> d_out (pos_out flat, then vel_out flat)
  dim3 g3N((N * 3 + 255) / 256);
  finalize_kernel<<<g3N, b256, 0, stream>>>(pos, pos2, x3, (float*)d_out, N);
}